// TopKGatingNetwork_55980603736498
// MI455X (gfx1250) — compile-verified
//
#include <hip/hip_runtime.h>
#include <math.h>

typedef __attribute__((ext_vector_type(2))) float v2f;
typedef __attribute__((ext_vector_type(8))) float v8f;

// Problem geometry (fixed by the reference: N=16384, D=4096, E=64, k=2)
#define TILE_D 4096
#define TILE_E 64
#define ROWS_PER_BLOCK 64          // 4 row-tiles of 16
#define LDS_STRIDE (TILE_E + 1)    // 65 floats: kills bank conflicts on the top-k scan

__global__ __launch_bounds__(256) void moe_router_topk2_kernel(
    const float* __restrict__ x,      // [N, D] row-major
    const float* __restrict__ W,      // [E, D] row-major (so B = W^T comes free)
    const float* __restrict__ bias,   // [E]
    float* __restrict__ gates,        // [N, E]
    float* __restrict__ topidx,       // [N, 2] (indices encoded as floats)
    int N)
{
    __shared__ float lds[ROWS_PER_BLOCK * LDS_STRIDE];
    __shared__ int   s_i0[ROWS_PER_BLOCK], s_i1[ROWS_PER_BLOCK];
    __shared__ float s_p0[ROWS_PER_BLOCK], s_p1[ROWS_PER_BLOCK];

    const int tid   = threadIdx.x;
    const int wave  = tid >> 5;          // 0..7
    const int lane  = tid & 31;
    const int rowTile = wave >> 1;       // 0..3 : which 16-row tile
    const int eHalf   = wave & 1;        // 0..1 : experts [0,32) or [32,64)
    const int n     = lane & 15;         // row-within-tile for A / expert-col for B
    const int khalf = lane >> 4;         // K-pair selector (ISA 16x4 f32 A/B layout)

    const int blockRow0 = blockIdx.x * ROWS_PER_BLOCK;
    const int waveRow0  = blockRow0 + rowTile * 16;
    const int eBase     = eHalf * 32;

    // A fragment stream: lane holds x[row = waveRow0+n, k0 + 2*khalf + {0,1}]
    const float* xA  = x + (size_t)(waveRow0 + n) * TILE_D + 2 * khalf;
    // B fragments: lane holds W[e, k0 + 2*khalf + {0,1}]  (B = W^T, same layout as A)
    const float* wB0 = W + (size_t)(eBase +      n) * TILE_D + 2 * khalf;
    const float* wB1 = W + (size_t)(eBase + 16 + n) * TILE_D + 2 * khalf;

    v8f acc0 = {};  // 16x16 f32 tile: experts [eBase, eBase+16)
    v8f acc1 = {};  // 16x16 f32 tile: experts [eBase+16, eBase+32)

    for (int k0 = 0; k0 < TILE_D; k0 += 32) {
        // Stream-ahead prefetch of this wave's x row. locality=3 -> WGP scope,
        // which (per prefetch-TH rules) fills all cache levels, not just GL2.
        // Speculative: dropped silently if it runs past the allocation.
        __builtin_prefetch(xA + k0 + 256, 0, 3);
#pragma unroll
        for (int kk = 0; kk < 32; kk += 4) {
            v2f a  = *(const v2f*)(xA  + k0 + kk);
            v2f b0 = *(const v2f*)(wB0 + k0 + kk);
            v2f b1 = *(const v2f*)(wB1 + k0 + kk);
            // D = A(16x4,f32) * B(4x16,f32) + C, full fp32 accumulate
            acc0 = __builtin_amdgcn_wmma_f32_16x16x4_f32(
                false, a, false, b0, (short)0, acc0, false, false);
            acc1 = __builtin_amdgcn_wmma_f32_16x16x4_f32(
                false, a, false, b1, (short)0, acc1, false, false);
        }
    }

    // Bias is a function of the expert (N) dimension only -> one scalar per lane per tile.
    const float bb0 = bias[eBase + n];
    const float bb1 = bias[eBase + 16 + n];

    // C/D layout: VGPR r holds local row r (lanes 0-15) / r+8 (lanes 16-31), col = lane&15.
    const int rlo = rowTile * 16 + 8 * khalf;
#pragma unroll
    for (int r = 0; r < 8; ++r) {
        lds[(rlo + r) * LDS_STRIDE + eBase +      n] = acc0[r] + bb0;
        lds[(rlo + r) * LDS_STRIDE + eBase + 16 + n] = acc1[r] + bb1;
    }
    __syncthreads();

    // ---- Phase 2: per-row top-2 + softmax (one thread per row) ----
    if (tid < ROWS_PER_BLOCK) {
        const float* row = &lds[tid * LDS_STRIDE];
        float v0 = -INFINITY, v1 = -INFINITY;
        int   i0 = 0, i1 = 0;
        for (int e = 0; e < TILE_E; ++e) {
            float v = row[e];
            if (v > v0)      { v1 = v0; i1 = i0; v0 = v; i0 = e; } // strict > : ties keep lowest index first
            else if (v > v1) { v1 = v;  i1 = e; }
        }
        float e1 = expf(v1 - v0);        // max-subtracted 2-way softmax
        float inv = 1.0f / (1.0f + e1);
        s_i0[tid] = i0; s_i1[tid] = i1;
        s_p0[tid] = inv; s_p1[tid] = e1 * inv;

        const int grow = blockRow0 + tid;
        topidx[(size_t)grow * 2 + 0] = (float)i0;
        topidx[(size_t)grow * 2 + 1] = (float)i1;
    }
    __syncthreads();

    // ---- Phase 3: cooperative coalesced gate write (zeros + 2 probs per row) ----
#pragma unroll
    for (int i = 0; i < 16; ++i) {
        int flat = i * 256 + tid;        // contiguous over [row, col] -> coalesced stores
        int r = flat >> 6;
        int c = flat & 63;
        float v = (c == s_i0[r]) ? s_p0[r] : ((c == s_i1[r]) ? s_p1[r] : 0.0f);
        gates[(size_t)(blockRow0 + r) * TILE_E + c] = v;
    }
}

extern "C" void kernel_launch(void* const* d_in, const int* in_sizes, int n_in,
                              void* d_out, int out_size, void* d_ws, size_t ws_size,
                              hipStream_t stream) {
    const float* x = (const float*)d_in[0];
    const float* W = (const float*)d_in[1];
    const float* b = (const float*)d_in[2];
    // Derive geometry: in_sizes = { N*D, E*D, E, 1 }
    const int E = in_sizes[2];
    const int D = in_sizes[1] / E;
    const int N = in_sizes[0] / D;

    float* gates  = (float*)d_out;                       // [N, E]
    float* topidx = (float*)d_out + (size_t)N * E;       // [N, 2] as floats

    dim3 block(256);
    dim3 grid(N / ROWS_PER_BLOCK);
    hipLaunchKernelGGL(moe_router_topk2_kernel, grid, block, 0, stream,
                       x, W, b, gates, topidx, N);
}